// DeformableAttn_46394236731947
// MI455X (gfx1250) — compile-verified
//
#include <hip/hip_runtime.h>

// MSDA constants (match reference): B=2, N=21760, C=256, L=4, Hh=8, P=4, Dh=32
#define CC 256

typedef float v2f __attribute__((ext_vector_type(2)));
typedef float v8f __attribute__((ext_vector_type(8)));

__device__ __forceinline__ void async_wait_all() {
#if __has_builtin(__builtin_amdgcn_s_wait_asynccnt)
  __builtin_amdgcn_s_wait_asynccnt(0);
#else
  asm volatile("s_wait_asynccnt 0x0" ::: "memory");
#endif
}

// ---------------------------------------------------------------------------
// GEMM: D[M, Ncols] = A[M, 256] @ W[256, Ncols] + bias[Ncols]
// Block: 256 threads = 8 waves. Tile: 128 rows x 64 cols.
// Each wave: one 16-row strip, four 16-col WMMA tiles (v_wmma_f32_16x16x4_f32).
//
// W is staged into LDS with the async Global->LDS engine
// (global_load_async_to_lds_b32, ASYNCcnt), K-pair interleaved so each B
// fragment {W[k][c], W[k+1][c]} is one aligned ds_load_b64:
//   LDS float2 index of (k, c) pair = (k>>1)*80 + c      (pad 80 => bank-safe:
//   lane-half kp offset = 160 floats = 32 banks; halves hit disjoint banks)
// M % 128 == 0 and Ncols % 64 == 0 (holds for all launches here).
// ---------------------------------------------------------------------------
__global__ __launch_bounds__(256) void gemm_wmma_f32(
    const float* __restrict__ A, const float* __restrict__ W,
    const float* __restrict__ bias, float* __restrict__ D,
    int Ncols)
{
  const int tid  = threadIdx.x;
  const int lane = tid & 31;
  const int wave = tid >> 5;
  const int m0 = blockIdx.x * 128 + wave * 16;
  const int n0 = blockIdx.y * 64;

  __shared__ float sB[64 * 80 * 2];   // 64 K-pairs x 80 float2 (40 KB)

  const int row   = lane & 15;          // M row within 16-row strip
  const int khalf = (lane >> 4) << 1;   // 0 for lanes 0-15, 2 for lanes 16-31
  const float* Arow = A + (size_t)(m0 + row) * CC;
  const v2f* sBp = (const v2f*)sB;

  v8f acc0 = {}, acc1 = {}, acc2 = {}, acc3 = {};

  for (int kc = 0; kc < CC; kc += 128) {
    __syncthreads();
    // Async-stage W[kc..kc+127, n0..n0+63] -> LDS, K-pair interleaved.
    // Each lane: own global address + own LDS address; DMA does the shuffle.
    for (int e = tid; e < 128 * 64; e += 256) {
      int k = e >> 6;                  // 0..127 (K within chunk)
      int c = e & 63;                  // 0..63  (column within tile)
      const float* gp = W + (size_t)(kc + k) * Ncols + (n0 + c);
      unsigned ldsa = (unsigned)(size_t)&sB[(((unsigned)(k >> 1) * 80u) +
                                            (unsigned)c) * 2u + (unsigned)(k & 1)];
      asm volatile("global_load_async_to_lds_b32 %0, %1, off"
                   :: "v"(ldsa), "v"(gp) : "memory");
    }
    async_wait_all();
    __syncthreads();

    #pragma unroll 4
    for (int k = 0; k < 128; k += 4) {
      // A fragment: 16x4 fp32; lanes 0-15 hold K={k,k+1}, lanes 16-31 K={k+2,k+3}
      v2f a = *(const v2f*)(Arow + kc + k + khalf);
      // B fragments: one contiguous float2 per tile, 8B-aligned ds_load_b64
      const v2f* frag = sBp + (size_t)((k >> 1) + (lane >> 4)) * 80 + (lane & 15);
      v2f b0 = frag[0];
      v2f b1 = frag[16];
      v2f b2 = frag[32];
      v2f b3 = frag[48];
      acc0 = __builtin_amdgcn_wmma_f32_16x16x4_f32(false, a, false, b0, (short)0, acc0, false, false);
      acc1 = __builtin_amdgcn_wmma_f32_16x16x4_f32(false, a, false, b1, (short)0, acc1, false, false);
      acc2 = __builtin_amdgcn_wmma_f32_16x16x4_f32(false, a, false, b2, (short)0, acc2, false, false);
      acc3 = __builtin_amdgcn_wmma_f32_16x16x4_f32(false, a, false, b3, (short)0, acc3, false, false);
    }
  }

  // Epilogue: C/D layout — VGPR r: lanes 0-15 -> M=r, lanes 16-31 -> M=r+8
  const int colg  = n0 + (lane & 15);
  const int rbase = (lane >> 4) * 8;
  const float bv0 = bias[colg +  0];
  const float bv1 = bias[colg + 16];
  const float bv2 = bias[colg + 32];
  const float bv3 = bias[colg + 48];
  #pragma unroll
  for (int r = 0; r < 8; ++r) {
    float* drow = D + (size_t)(m0 + rbase + r) * Ncols;
    drow[colg +  0] = acc0[r] + bv0;
    drow[colg + 16] = acc1[r] + bv1;
    drow[colg + 32] = acc2[r] + bv2;
    drow[colg + 48] = acc3[r] + bv3;
  }
}

// ---------------------------------------------------------------------------
// MSDA core: one 256-thread block per (b,n).
// thread tid -> head h = tid>>5, channel d = tid&31 (value channel h*32+d).
// Softmax over the 32 logits per head; 16 bilinear taps per thread.
// value tensor (44.6 MB) is fully L2-resident (192 MB L2), so the scattered
// corner loads run at L2 bandwidth; all 32 lanes of a head share loc/weights
// so each corner load is one coalesced 128B line.
// ---------------------------------------------------------------------------
__global__ __launch_bounds__(256) void msda_kernel(
    const float* __restrict__ offv,    // [B*N, 256]  (Hh,L,P,2)
    const float* __restrict__ attnv,   // [B*N, 128]  (Hh,L*P) logits
    const float* __restrict__ priors,  // [B*N, L, 2]
    const float* __restrict__ value,   // [B*S, 256]
    const int*   __restrict__ shapes,  // [L,2] = (H,W)
    const int*   __restrict__ starts,  // [L]
    float*       __restrict__ weighted,// [B*N, 256]
    int Nq, int S)
{
  __shared__ float s_off[256];
  __shared__ float s_aw[128];
  __shared__ float s_pri[8];
  __shared__ int   s_hw[8];
  __shared__ int   s_s0[4];

  const int bid = blockIdx.x;          // b*Nq + n
  const int tid = threadIdx.x;
  const int b   = bid / Nq;

  s_off[tid] = offv[(size_t)bid * 256 + tid];
  if (tid < 128) s_aw[tid]  = attnv[(size_t)bid * 128 + tid];
  if (tid < 8)   s_pri[tid] = priors[(size_t)bid * 8 + tid];
  if (tid < 8)   s_hw[tid]  = shapes[tid];
  if (tid < 4)   s_s0[tid]  = starts[tid];
  __syncthreads();

  if (tid < 8) {  // per-head softmax over 32 logits
    float mx = -1e30f;
    #pragma unroll
    for (int i = 0; i < 32; ++i) mx = fmaxf(mx, s_aw[tid * 32 + i]);
    float sum = 0.f;
    #pragma unroll
    for (int i = 0; i < 32; ++i) {
      float e = expf(s_aw[tid * 32 + i] - mx);
      s_aw[tid * 32 + i] = e;
      sum += e;
    }
    float inv = 1.0f / sum;
    #pragma unroll
    for (int i = 0; i < 32; ++i) s_aw[tid * 32 + i] *= inv;
  }
  __syncthreads();

  const int h = tid >> 5;
  const int d = tid & 31;
  const float* vbase = value + (size_t)b * S * 256 + (h * 32 + d);
  float acc = 0.f;

  #pragma unroll
  for (int l = 0; l < 4; ++l) {
    const int   Hl = s_hw[2 * l], Wl = s_hw[2 * l + 1];
    const int   s0 = s_s0[l];
    const float px = s_pri[2 * l], py = s_pri[2 * l + 1];
    const float fW = (float)Wl, fH = (float)Hl;
    #pragma unroll
    for (int p = 0; p < 4; ++p) {
      const int oi = ((h * 4 + l) * 4 + p) * 2;
      // loc = prior + off/(W,H); sample coord = loc*(W,H) - 0.5
      float lx = px * fW + s_off[oi]     - 0.5f;
      float ly = py * fH + s_off[oi + 1] - 0.5f;
      float x0f = floorf(lx), y0f = floorf(ly);
      float wx1 = lx - x0f, wy1 = ly - y0f;
      float wx0 = 1.f - wx1, wy0 = 1.f - wy1;
      int x0 = (int)x0f, y0 = (int)y0f;
      float aw = s_aw[h * 32 + l * 4 + p];

      float v00 = 0.f, v10 = 0.f, v01 = 0.f, v11 = 0.f;
      if (y0 >= 0 && y0 < Hl) {
        const float* rp = vbase + (size_t)(s0 + y0 * Wl) * 256;
        if (x0 >= 0 && x0 < Wl)         v00 = rp[(size_t)x0 * 256];
        if (x0 + 1 >= 0 && x0 + 1 < Wl) v10 = rp[(size_t)(x0 + 1) * 256];
      }
      if (y0 + 1 >= 0 && y0 + 1 < Hl) {
        const float* rp = vbase + (size_t)(s0 + (y0 + 1) * Wl) * 256;
        if (x0 >= 0 && x0 < Wl)         v01 = rp[(size_t)x0 * 256];
        if (x0 + 1 >= 0 && x0 + 1 < Wl) v11 = rp[(size_t)(x0 + 1) * 256];
      }
      acc += aw * (wy0 * (wx0 * v00 + wx1 * v10) +
                   wy1 * (wx0 * v01 + wx1 * v11));
    }
  }
  weighted[(size_t)bid * 256 + tid] = acc;
}

// ---------------------------------------------------------------------------
extern "C" void kernel_launch(void* const* d_in, const int* in_sizes, int n_in,
                              void* d_out, int out_size, void* d_ws, size_t ws_size,
                              hipStream_t stream)
{
  (void)in_sizes; (void)n_in; (void)out_size; (void)ws_size;

  const float* in_feats = (const float*)d_in[0];   // [B,N,256]
  const float* priors   = (const float*)d_in[1];   // [B,N,4,2]
  const float* samp     = (const float*)d_in[2];   // [B,S,256]
  const int*   shapes   = (const int*)  d_in[3];   // [4,2]
  const int*   starts   = (const int*)  d_in[4];   // [4]
  const float* W_off    = (const float*)d_in[5];
  const float* b_off    = (const float*)d_in[6];
  const float* W_attn   = (const float*)d_in[7];
  const float* b_attn   = (const float*)d_in[8];
  const float* W_val    = (const float*)d_in[9];
  const float* b_val    = (const float*)d_in[10];
  const float* W_out    = (const float*)d_in[11];
  const float* b_out    = (const float*)d_in[12];
  float* out = (float*)d_out;

  const int Bb = 2, Nq = 21760, S = 21760;
  const int M  = Bb * Nq;                 // 43520 rows (== Bb*S)

  float* ws    = (float*)d_ws;
  float* vVal  = ws;                                  // M*256
  float* vOff  = vVal + (size_t)M * 256;              // M*256
  float* vAttn = vOff + (size_t)M * 256;              // M*128
  float* vWgt  = vAttn + (size_t)M * 128;             // M*256

  dim3 blk(256);
  // value = sample_feats @ W_val + b_val
  gemm_wmma_f32<<<dim3(M / 128, 4), blk, 0, stream>>>(samp,     W_val,  b_val,  vVal,  256);
  // off   = in_feats @ W_off + b_off
  gemm_wmma_f32<<<dim3(M / 128, 4), blk, 0, stream>>>(in_feats, W_off,  b_off,  vOff,  256);
  // attn  = in_feats @ W_attn + b_attn
  gemm_wmma_f32<<<dim3(M / 128, 2), blk, 0, stream>>>(in_feats, W_attn, b_attn, vAttn, 128);
  // weighted = softmax + bilinear gather-accumulate
  msda_kernel<<<dim3(M), blk, 0, stream>>>(vOff, vAttn, priors, vVal, shapes, starts, vWgt, Nq, S);
  // out = weighted @ W_out + b_out
  gemm_wmma_f32<<<dim3(M / 128, 4), blk, 0, stream>>>(vWgt, W_out, b_out, out, 256);
}